// MyLoss_7971459302100
// MI455X (gfx1250) — compile-verified
//
#include <hip/hip_runtime.h>
#include <hip/hip_bf16.h>

// Loss decomposition (all sums over the whole batch, then /B at the end):
//   noobj_all  = 0.5 * sum_{all cells} (c3^2 + c7^2)                (kernel A)
//   per-label  = 5*coord + (big-conf)^2 + 0.5*(small-unconf)^2
//                - 0.5*(c3^2+c7^2) for each DISTINCT object cell    (kernel B)
//   final      = (sum of all block partials) / B                    (kernel C, WMMA reduce)
//
// Roofline: ~148 MB of traffic @ 23.3 TB/s => ~6.4 us floor; kernel A carries
// ~128 MB of it, so it is written for maximum memory-level parallelism
// (8 independent global_load_b128 in flight per loop iteration).

#define LAMBDA_COORD 5.0f
#define LAMBDA_NOOBJ 0.5f

typedef float v2f __attribute__((ext_vector_type(2)));
typedef float v8f __attribute__((ext_vector_type(8)));

__device__ __forceinline__ float bbox_iou(float px, float py, float pr,
                                          float gx, float gy, float gr) {
    float l1 = px - pr, r1 = px + pr, t1 = py - pr, b1 = py + pr;
    float l2 = gx - gr, r2 = gx + gr, t2 = gy - gr, b2 = gy + gr;
    float iw = fmaxf(fminf(r1, r2) - fmaxf(l1, l2), 0.0f);
    float ih = fmaxf(fminf(b1, b2) - fmaxf(t1, t2), 0.0f);
    float inter = iw * ih;
    float a1 = (r1 - l1) * (b1 - t1);
    float a2 = (r2 - l2) * (b2 - t2);
    return inter / (a1 + a2 - inter + 1e-9f);
}

// Deterministic block reduction: wave32 shfl_xor tree, then LDS across waves.
__device__ __forceinline__ float block_reduce(float v, float* sm) {
    #pragma unroll
    for (int m = 16; m >= 1; m >>= 1) v += __shfl_xor(v, m, 32);
    int wid = threadIdx.x >> 5;
    int lid = threadIdx.x & 31;
    if (lid == 0) sm[wid] = v;
    __syncthreads();
    int nw = blockDim.x >> 5;
    if (wid == 0) {
        v = (lid < nw) ? sm[lid] : 0.0f;
        #pragma unroll
        for (int m = 16; m >= 1; m >>= 1) v += __shfl_xor(v, m, 32);
    }
    return v;  // valid in wave 0
}

// ---------- Kernel A: streaming no-obj sum over ALL cells (memory-bound) ----
// pred viewed as float4: cell i = f4[2i], f4[2i+1]; channels 3,7 are the .w's.
// Stride loop unrolled x4: 8 independent b128 loads issued before any use.
__global__ void yolo_noobj_all(const float4* __restrict__ p4,
                               float* __restrict__ partials, int ncells) {
    int tid = (int)(blockIdx.x * blockDim.x + threadIdx.x);
    int stride = (int)(gridDim.x * blockDim.x);
    float acc = 0.0f;

    int i = tid;
    for (; i + 3 * stride < ncells; i += 4 * stride) {
        // prefetch the next unrolled chunk (global_prefetch_b8)
        if (i + 4 * stride < ncells)
            __builtin_prefetch((const void*)(p4 + 2u * (size_t)(i + 4 * stride)), 0, 1);
        float4 a0 = p4[2u * (size_t)i];
        float4 b0 = p4[2u * (size_t)i + 1u];
        float4 a1 = p4[2u * (size_t)(i + stride)];
        float4 b1 = p4[2u * (size_t)(i + stride) + 1u];
        float4 a2 = p4[2u * (size_t)(i + 2 * stride)];
        float4 b2 = p4[2u * (size_t)(i + 2 * stride) + 1u];
        float4 a3 = p4[2u * (size_t)(i + 3 * stride)];
        float4 b3 = p4[2u * (size_t)(i + 3 * stride) + 1u];
        acc = fmaf(a0.w, a0.w, acc); acc = fmaf(b0.w, b0.w, acc);
        acc = fmaf(a1.w, a1.w, acc); acc = fmaf(b1.w, b1.w, acc);
        acc = fmaf(a2.w, a2.w, acc); acc = fmaf(b2.w, b2.w, acc);
        acc = fmaf(a3.w, a3.w, acc); acc = fmaf(b3.w, b3.w, acc);
    }
    for (; i < ncells; i += stride) {   // tail (empty when ncells % (4*stride) == 0)
        float4 a = p4[2u * (size_t)i];
        float4 b = p4[2u * (size_t)i + 1u];
        acc = fmaf(a.w, a.w, acc);
        acc = fmaf(b.w, b.w, acc);
    }

    acc *= LAMBDA_NOOBJ;
    __shared__ float sm[8];
    float r = block_reduce(acc, sm);
    if (threadIdx.x == 0) partials[blockIdx.x] = r;
}

// ---------- Kernel B: per-label gather + loss, minus obj-cell noobj terms ----
// One thread per batch element; pred (128 MB) fits the 192 MB L2, so these
// gathers mostly hit L2 after kernel A streamed it through.
__global__ void yolo_labels(const float* __restrict__ pred,
                            const int* __restrict__ rc,
                            const float* __restrict__ box,
                            float* __restrict__ partials, int Bn) {
    int b = (int)(blockIdx.x * blockDim.x + threadIdx.x);
    float acc = 0.0f;
    if (b < Bn) {
        const int*   rcb  = rc  + (size_t)b * 32;   // [16][2]
        const float* boxb = box + (size_t)b * 48;   // [16][3]
        const float* pb   = pred + (size_t)b * 512; // [8][8][8]
        unsigned long long seen = 0ull;
        #pragma unroll
        for (int l = 0; l < 16; ++l) {
            int r = rcb[2 * l], c = rcb[2 * l + 1];
            float gx = boxb[3 * l], gy = boxb[3 * l + 1], gr = boxb[3 * l + 2];
            int cellidx = r * 8 + c;
            const float4* cell = (const float4*)(pb + (size_t)cellidx * 8);
            float4 lo = cell[0];  // ch 0..3
            float4 hi = cell[1];  // ch 4..7
            float iou1 = bbox_iou(lo.x, lo.y, lo.z, gx, gy, gr);
            float iou2 = bbox_iou(hi.x, hi.y, hi.z, gx, gy, gr);
            bool sw = iou2 > iou1;
            float cx      = sw ? hi.x : lo.x;
            float cy      = sw ? hi.y : lo.y;
            float cr      = sw ? hi.z : lo.z;
            float conf    = sw ? hi.w : lo.w;
            float unconf  = sw ? lo.w : hi.w;
            float bigger  = sw ? iou2 : iou1;
            float smaller = sw ? iou1 : iou2;
            float dx = cx - gx, dy = cy - gy, dr = cr - gr;
            acc += LAMBDA_COORD * (dx * dx + dy * dy + dr * dr);
            float dc = bigger - conf;
            float du = smaller - unconf;
            acc += dc * dc + LAMBDA_NOOBJ * (du * du);
            // subtract this cell's contribution to the all-cells noobj sum,
            // exactly once per distinct cell (duplicates collapse, as in ref)
            unsigned long long bit = 1ull << cellidx;
            if (!(seen & bit)) {
                seen |= bit;
                acc -= LAMBDA_NOOBJ * (lo.w * lo.w + hi.w * hi.w);
            }
        }
    }
    __shared__ float sm[8];
    float s = block_reduce(acc, sm);
    if (threadIdx.x == 0) partials[blockIdx.x] = s;
}

// ---------- Kernel C: single-wave final reduce using V_WMMA_F32_16X16X4_F32 -
// 64 fp32 accumulators live in A (2 VGPRs x 32 lanes). B = ones:
//   D[m][n] = A[m][0]+A[m][1]+A[m][2]+A[m][3]  -> 16 row sums in ONE wmma.
// D VGPR r holds S_r (lanes 0-15) and S_{8+r} (lanes 16-31); finish with
// 7 VALU adds + one shuffle. Exact fp32 math, fixed order -> deterministic.
__global__ void yolo_final(const float* __restrict__ partials, int n,
                           float* __restrict__ out, float invB) {
    int lane = (int)threadIdx.x;  // launched with exactly 32 threads
    float x = 0.0f, y = 0.0f;
    for (int i = lane; i < n; i += 64)       x += partials[i];
    for (int i = lane + 32; i < n; i += 64)  y += partials[i];
    v2f a;  a[0] = x;    a[1] = y;      // A[m][k]: K=0/1 (lanes 0-15), K=2/3 (lanes 16-31)
    v2f bo; bo[0] = 1.0f; bo[1] = 1.0f; // B = all ones
    v8f cz = {};
    v8f d = __builtin_amdgcn_wmma_f32_16x16x4_f32(
        false, a, false, bo, (short)0, cz, false, false);
    float t = d[0] + d[1] + d[2] + d[3] + d[4] + d[5] + d[6] + d[7];
    // lanes 0-15 hold sum(S_0..S_7); lanes 16-31 hold sum(S_8..S_15)
    float other = __shfl(t, 16, 32);
    if (lane == 0) out[0] = (t + other) * invB;
}

extern "C" void kernel_launch(void* const* d_in, const int* in_sizes, int n_in,
                              void* d_out, int out_size, void* d_ws, size_t ws_size,
                              hipStream_t stream) {
    const float* pred = (const float*)d_in[0];
    const int*   rc   = (const int*)d_in[1];
    const float* box  = (const float*)d_in[2];
    float*       out  = (float*)d_out;
    float*       ws   = (float*)d_ws;

    int B = in_sizes[0] / 512;         // pred is [B, 8, 8, 8]
    int ncells = B * 64;

    const int blkA = 256;
    const int gridA = 2048;            // streaming pass: 8 cells/thread at B=65536
    const int blkB = 256;
    const int gridB = (B + blkB - 1) / blkB;
    const int nPartials = gridA + gridB;

    float* partA = ws;
    float* partB = ws + gridA;

    yolo_noobj_all<<<gridA, blkA, 0, stream>>>((const float4*)pred, partA, ncells);
    yolo_labels<<<gridB, blkB, 0, stream>>>(pred, rc, box, partB, B);
    yolo_final<<<1, 32, 0, stream>>>(ws, nPartials, out, 1.0f / (float)B);
}